// SAGEConv_53884659695845
// MI455X (gfx1250) — compile-verified
//
#include <hip/hip_runtime.h>

typedef __attribute__((ext_vector_type(2))) float v2f;
typedef __attribute__((ext_vector_type(8))) float v8f;

#define D_IN   96
#define D_K    192
#define D_OUT  128
#define BN_EPS 1e-5f
// 16 x 192 tile padded to stride 196 floats:
//  - row base = 784 B -> 16B aligned (legal B128 LDS writes for async DMA)
//  - 196 % 64 == 4 -> a-frag reads cover 64 distinct banks (conflict-free)
#define LDSTRIDE 196

// ---------------- K1: zero workspace ----------------
__global__ void zero_ws(float* __restrict__ p, long n) {
  long i = blockIdx.x * (long)blockDim.x + threadIdx.x;
  if (i < n) p[i] = 0.0f;
}

// ---------------- K2: edge scatter-sum (one wave32 per edge) ----------------
__global__ void scatter_sum(const float* __restrict__ x,
                            const long long* __restrict__ edges,
                            float* __restrict__ nsum,
                            float* __restrict__ cnt,
                            int E) {
  int warp = (int)((blockIdx.x * (long)blockDim.x + threadIdx.x) >> 5);
  int lane = threadIdx.x & 31;
  if (warp >= E) return;
  long long src = edges[2 * (long)warp];
  long long dst = edges[2 * (long)warp + 1];
  const float* xs = x + src * D_IN;
  float* ns = nsum + dst * D_IN;
#pragma unroll
  for (int i = 0; i < 3; ++i) {          // 96 features / 32 lanes
    int f = lane + 32 * i;
    atomicAdd(&ns[f], xs[f]);
  }
  if (lane == 0) atomicAdd(&cnt[dst], 1.0f);
}

// ---------------- K3: fused [x | mean] GEMM via V_WMMA_F32_16X16X4_F32 ----------------
__global__ void __launch_bounds__(256)
gemm_wmma(const float* __restrict__ x,
          const float* __restrict__ nsum,
          const float* __restrict__ cnt,
          const float* __restrict__ W,       // (D_OUT, D_K) row-major
          const float* __restrict__ bias,
          float* __restrict__ out_pre,       // (N, D_OUT), pre-BN
          float* __restrict__ colsum,
          float* __restrict__ colsumsq,
          int N) {
  __shared__ float As[16 * LDSTRIDE];
  __shared__ float rc[16];

  const int row0 = blockIdx.x * 16;
  const int tid  = threadIdx.x;

  if (tid < 16) {
    int n = row0 + tid;
    float c = (n < N) ? cnt[n] : 1.0f;
    rc[tid] = 1.0f / fmaxf(c, 1.0f);
  }
  __syncthreads();

  // --- x half (cols 0..95): async DMA global -> LDS, 16B granules ---
  // 16 rows x 24 float4 = 384 transfers; each row of x is 384 B, 16B aligned.
  for (int idx = tid; idx < 16 * 24; idx += blockDim.x) {
    int r  = idx / 24;
    int c4 = (idx % 24) * 4;
    int n  = row0 + r;
    if (n < N) {
      unsigned     lds_off = (unsigned)(uintptr_t)&As[r * LDSTRIDE + c4];
      const float* gaddr   = x + (long)n * D_IN + c4;
      asm volatile("global_load_async_to_lds_b128 %0, %1, off"
                   :: "v"(lds_off), "v"(gaddr)
                   : "memory");
    } else {
      As[r * LDSTRIDE + c4 + 0] = 0.0f;
      As[r * LDSTRIDE + c4 + 1] = 0.0f;
      As[r * LDSTRIDE + c4 + 2] = 0.0f;
      As[r * LDSTRIDE + c4 + 3] = 0.0f;
    }
  }

  // --- mean half (cols 96..191): needs the 1/max(cnt,1) scale -> VALU path ---
  for (int idx = tid; idx < 16 * D_IN; idx += blockDim.x) {
    int r = idx / D_IN, c = idx % D_IN;
    int n = row0 + r;
    As[r * LDSTRIDE + D_IN + c] =
        (n < N) ? nsum[(long)n * D_IN + c] * rc[r] : 0.0f;
  }

  // drain this wave's async DMA before the tile barrier
  asm volatile("s_wait_asynccnt 0x0" ::: "memory");
  __syncthreads();

  const int wave = tid >> 5;           // 8 waves -> 8 column tiles of 16
  const int lane = tid & 31;
  const int half = lane >> 4;          // K sub-pair select (A/B frag layout)
  const int m    = lane & 15;
  const int col  = wave * 16 + m;

  const float* wp = W + (long)col * D_K;
  __builtin_prefetch(wp, 0, 3);                       // global_prefetch_b8
  __builtin_prefetch(wp + 128, 0, 3);

  v8f acc = {};
#pragma unroll 4
  for (int k0 = 0; k0 < D_K; k0 += 4) {
    v2f a, bf;
    // A 16x4 f32 frag: lane<16 -> K = k0,k0+1 of row m ; lane>=16 -> K = k0+2,k0+3
    a.x  = As[m * LDSTRIDE + k0 + 2 * half];
    a.y  = As[m * LDSTRIDE + k0 + 2 * half + 1];
    // B 4x16 f32 frag: B[k][col] = W[col][k]
    bf.x = wp[k0 + 2 * half];
    bf.y = wp[k0 + 2 * half + 1];
    acc = __builtin_amdgcn_wmma_f32_16x16x4_f32(false, a, false, bf,
                                                (short)0, acc, false, false);
  }

  // D layout: VGPR j holds M = j + 8*half, N = m. Add bias, store, fold column stats.
  const float bcol = bias[col];
  float s = 0.0f, s2 = 0.0f;
  if (row0 + 16 <= N) {
    // full tile (always true for N % 16 == 0): unguarded store clause
#pragma unroll
    for (int j = 0; j < 8; ++j) {
      int n = row0 + half * 8 + j;
      float v = acc[j] + bcol;
      out_pre[(long)n * D_OUT + col] = v;
      s  += v;
      s2 += v * v;
    }
  } else {
#pragma unroll
    for (int j = 0; j < 8; ++j) {
      int n = row0 + half * 8 + j;
      float v = acc[j] + bcol;
      if (n < N) {
        out_pre[(long)n * D_OUT + col] = v;
        s  += v;
        s2 += v * v;
      }
    }
  }
  atomicAdd(&colsum[col],   s);
  atomicAdd(&colsumsq[col], s2);
}

// ---------------- K4: BatchNorm (axis 0) + ReLU, in place ----------------
__global__ void bn_relu(float* __restrict__ out,
                        const float* __restrict__ colsum,
                        const float* __restrict__ colsumsq,
                        const float* __restrict__ gamma,
                        const float* __restrict__ beta,
                        int N) {
  long t = blockIdx.x * (long)blockDim.x + threadIdx.x;
  long total = (long)N * D_OUT;
  if (t >= total) return;
  int o = (int)(t % D_OUT);
  float invN = 1.0f / (float)N;
  float mean = colsum[o] * invN;
  float var  = colsumsq[o] * invN - mean * mean;
  float v = (out[t] - mean) * rsqrtf(var + BN_EPS) * gamma[o] + beta[o];
  out[t] = fmaxf(v, 0.0f);
}

// ---------------- launch ----------------
extern "C" void kernel_launch(void* const* d_in, const int* in_sizes, int n_in,
                              void* d_out, int out_size, void* d_ws, size_t ws_size,
                              hipStream_t stream) {
  const float*     x     = (const float*)d_in[0];
  const long long* edges = (const long long*)d_in[1];   // int64 (E,2)
  // d_in[2] = num_nodes scalar (derive N from sizes instead)
  const float*     W     = (const float*)d_in[3];
  const float*     b     = (const float*)d_in[4];
  const float*     gamma = (const float*)d_in[5];
  const float*     beta  = (const float*)d_in[6];

  const int N = in_sizes[0] / D_IN;
  const int E = in_sizes[1] / 2;

  float* ws       = (float*)d_ws;
  float* nsum     = ws;                         // N * 96
  float* cnt      = nsum + (long)N * D_IN;      // N
  float* colsum   = cnt + N;                    // 128
  float* colsumsq = colsum + D_OUT;             // 128

  long zn = (long)N * D_IN + N + 2 * D_OUT;
  zero_ws<<<(int)((zn + 255) / 256), 256, 0, stream>>>(ws, zn);

  long sthreads = (long)E * 32;
  scatter_sum<<<(int)((sthreads + 255) / 256), 256, 0, stream>>>(x, edges, nsum, cnt, E);

  int ntiles = (N + 15) / 16;
  gemm_wmma<<<ntiles, 256, 0, stream>>>(x, nsum, cnt, W, b,
                                        (float*)d_out, colsum, colsumsq, N);

  long total = (long)N * D_OUT;
  bn_relu<<<(int)((total + 255) / 256), 256, 0, stream>>>((float*)d_out, colsum, colsumsq,
                                                          gamma, beta, N);
}